// RandomFlip_45999099740646
// MI455X (gfx1250) — compile-verified
//
#include <hip/hip_runtime.h>
#include <cstdint>
#include <cstddef>

// Native clang vectors (HIP float4 is a class, unusable with these builtins).
typedef __attribute__((ext_vector_type(4))) float v4f;
typedef __attribute__((ext_vector_type(4))) int   v4i;   // param type of async-LDS builtins

// Only enable the async path in the device pass, and only if the toolchain
// declares the gfx1250 async-LDS builtins. Host pass / older toolchains fall
// back to the plain NT b128 copy body inside the same kernel symbol.
#if defined(__HIP_DEVICE_COMPILE__) && defined(__has_builtin)
#if __has_builtin(__builtin_amdgcn_global_load_async_to_lds_b128) && \
    __has_builtin(__builtin_amdgcn_global_store_async_from_lds_b128) && \
    __has_builtin(__builtin_amdgcn_s_wait_asynccnt)
#define HAVE_ASYNC_LDS 1
#endif
#endif
#ifndef HAVE_ASYNC_LDS
#define HAVE_ASYNC_LDS 0
#endif

// ---------------------------------------------------------------------------
// Compile-time replication of jax.random.uniform(jax.random.key(42), ()):
// threefry2x32 with key (0,42), counter (0,0); bits -> [0,1) float.
// The flip decision is therefore a compile-time constant, exactly as in the
// reference (the JAX trace bakes the same constant into the cond).
// ---------------------------------------------------------------------------
constexpr uint32_t rotl_(uint32_t x, int r) { return (x << r) | (x >> (32 - r)); }

constexpr float jax_uniform_key42() {
  const uint32_t k0 = 0u, k1 = 42u;
  const uint32_t ks2 = 0x1BD11BDAu ^ k0 ^ k1;
  uint32_t x0 = k0, x1 = k1;            // counter words are (0,0)
  const int RA[4] = {13, 15, 26, 6};
  const int RB[4] = {17, 29, 16, 24};
  for (int i = 0; i < 4; ++i) { x0 += x1; x1 = rotl_(x1, RA[i]) ^ x0; }
  x0 += k1;  x1 += ks2 + 1u;
  for (int i = 0; i < 4; ++i) { x0 += x1; x1 = rotl_(x1, RB[i]) ^ x0; }
  x0 += ks2; x1 += k0 + 2u;
  for (int i = 0; i < 4; ++i) { x0 += x1; x1 = rotl_(x1, RA[i]) ^ x0; }
  x0 += k0;  x1 += k1 + 3u;
  for (int i = 0; i < 4; ++i) { x0 += x1; x1 = rotl_(x1, RB[i]) ^ x0; }
  x0 += k1;  x1 += ks2 + 4u;
  for (int i = 0; i < 4; ++i) { x0 += x1; x1 = rotl_(x1, RA[i]) ^ x0; }
  x0 += ks2; x1 += k0 + 5u;
  const uint32_t f = (x0 >> 9) | 0x3F800000u;
  float v = __builtin_bit_cast(float, f) - 1.0f;
  return v < 0.0f ? 0.0f : v;  // lax.max(minval, ...)
}

constexpr bool DO_FLIP = (jax_uniform_key42() <= 0.5f);

// ---------------------------------------------------------------------------
// Geometry constants (fixed by the reference)
// ---------------------------------------------------------------------------
constexpr unsigned ROWS      = 32u * 640u;   // N*H = 20480 image rows
constexpr unsigned ROW_FLOAT = 640u * 3u;    // 1920 floats per row
constexpr unsigned CHUNKS    = 160u;         // 640 px / 4 px per chunk
constexpr unsigned NIMG      = 32u * 640u * 640u * 3u;  // 39,321,600 floats
constexpr unsigned NIMG4     = NIMG / 4u;               // 9,830,400 float4
constexpr unsigned NBOX4     = 32u * 50u;    // 1600 float4 boxes
constexpr unsigned NSCAL     = 32u * 50u;    // 1600 distances / classes

// ---------------------------------------------------------------------------
// Horizontal flip: out[n,y,x,c] = in[n,y,639-x,c].
// One thread = 4 output pixels = 48 B. Both sides are 16B-aligned b128 ops.
// Streaming data (315 MB > 192 MB L2, touched once) -> non-temporal hints.
// ---------------------------------------------------------------------------
__global__ __launch_bounds__(256) void img_flip_kernel(
    const float* __restrict__ in, float* __restrict__ out) {
  const unsigned t = blockIdx.x * 256u + threadIdx.x;
  if (t >= ROWS * CHUNKS) return;
  const unsigned cx  = t % CHUNKS;
  const unsigned row = t / CHUNKS;
  const size_t rowbase = (size_t)row * ROW_FLOAT;

  // input pixels (636-4cx)..(639-4cx); 12 floats, 16B aligned
  const v4f* src = (const v4f*)(in + rowbase + (size_t)(1908u - 12u * cx));
  const v4f a = __builtin_nontemporal_load(src + 0);  // f0..f3
  const v4f b = __builtin_nontemporal_load(src + 1);  // f4..f7
  const v4f c = __builtin_nontemporal_load(src + 2);  // f8..f11

  // reverse pixel order, keep channel order:
  // out = [f9 f10 f11 | f6 f7 f8 | f3 f4 f5 | f0 f1 f2]
  const v4f o0 = {c.y, c.z, c.w, b.z};
  const v4f o1 = {b.w, c.x, a.w, b.x};
  const v4f o2 = {b.y, a.x, a.y, a.z};

  v4f* dst = (v4f*)(out + rowbase + (size_t)(12u * cx));
  __builtin_nontemporal_store(o0, dst + 0);
  __builtin_nontemporal_store(o1, dst + 1);
  __builtin_nontemporal_store(o2, dst + 2);
}

// ---------------------------------------------------------------------------
// Identity copy path (not taken for key 42, where DO_FLIP is true).
// Preferred: CDNA5 async data-mover, HBM -> LDS -> HBM with no VGPR round
// trip, tracked on ASYNCcnt. Each block stages 16 KB in LDS (4 slots x 256
// lanes x 16 B); sizes divide exactly so EXEC is all-ones for every async op.
// Fallback body (host pass / missing builtins): plain NT b128 stream.
// ---------------------------------------------------------------------------
__global__ __launch_bounds__(256) void img_copy_kernel(
    const v4f* __restrict__ in, v4f* __restrict__ out, unsigned n4) {
#if HAVE_ASYNC_LDS
  __shared__ v4f lbuf[1024];  // 16 KB staging
  const unsigned tid  = threadIdx.x;
  const unsigned base = blockIdx.x * 1024u + tid;
#pragma unroll
  for (unsigned k = 0; k < 4; ++k) {
    const unsigned i = base + k * 256u;
    __builtin_amdgcn_global_load_async_to_lds_b128(
        (v4i*)(in + i), (v4i*)&lbuf[tid + k * 256u],
        /*offset=*/0, /*cpol=NT*/1);
  }
  __builtin_amdgcn_s_wait_asynccnt(0);  // LDS slots now hold the tile
#pragma unroll
  for (unsigned k = 0; k < 4; ++k) {
    const unsigned i = base + k * 256u;
    __builtin_amdgcn_global_store_async_from_lds_b128(
        (v4i*)(out + i), (v4i*)&lbuf[tid + k * 256u],
        /*offset=*/0, /*cpol=NT*/1);
  }
  // S_ENDPGM performs an implicit wait-idle, covering the outstanding stores.
  (void)n4;
#else
  const unsigned base = blockIdx.x * 1024u + threadIdx.x;
#pragma unroll
  for (int k = 0; k < 4; ++k) {
    const unsigned i = base + (unsigned)k * 256u;
    if (i < n4) {
      __builtin_nontemporal_store(__builtin_nontemporal_load(in + i), out + i);
    }
  }
#endif
}

// ---------------------------------------------------------------------------
// Tail: boxes (affine if flipped), distances copy, classes int32 -> f32
// (JAX promotes int32 to float32 when the outputs are flattened together).
// ---------------------------------------------------------------------------
__global__ __launch_bounds__(256) void tail_kernel(
    const v4f* __restrict__ boxes, const float* __restrict__ dist,
    const int* __restrict__ cls, v4f* __restrict__ obox,
    float* __restrict__ odist, float* __restrict__ ocls, int do_flip) {
  const unsigned t = blockIdx.x * 256u + threadIdx.x;
  if (t < NBOX4) {
    v4f b = boxes[t];
    if (do_flip) {
      v4f r = {640.0f - b.x, b.y, 640.0f - b.z, b.w};
      b = r;
    }
    obox[t] = b;
  } else if (t < NBOX4 + NSCAL) {
    const unsigned i = t - NBOX4;
    odist[i] = dist[i];
  } else if (t < NBOX4 + 2u * NSCAL) {
    const unsigned i = t - (NBOX4 + NSCAL);
    ocls[i] = (float)cls[i];
  }
}

extern "C" void kernel_launch(void* const* d_in, const int* in_sizes, int n_in,
                              void* d_out, int out_size, void* d_ws, size_t ws_size,
                              hipStream_t stream) {
  (void)in_sizes; (void)n_in; (void)out_size; (void)d_ws; (void)ws_size;

  const float* images = (const float*)d_in[0];
  const float* boxes  = (const float*)d_in[1];
  const float* dist   = (const float*)d_in[2];
  const int*   cls    = (const int*)d_in[3];

  float* out   = (float*)d_out;
  float* oimg  = out;                       // 39,321,600 floats
  float* obox  = out + NIMG;                //      6,400 floats
  float* odist = obox + 4u * NBOX4;         //      1,600 floats
  float* ocls  = odist + NSCAL;             //      1,600 floats

  if (DO_FLIP) {
    const unsigned nthreads = ROWS * CHUNKS;            // 3,276,800
    img_flip_kernel<<<nthreads / 256u, 256, 0, stream>>>(images, oimg);
  } else {
    const unsigned nblocks = NIMG4 / 1024u;             // 9,600
    img_copy_kernel<<<nblocks, 256, 0, stream>>>(
        (const v4f*)images, (v4f*)oimg, NIMG4);
  }

  const unsigned tail_threads = NBOX4 + 2u * NSCAL;     // 4,800
  tail_kernel<<<(tail_threads + 255u) / 256u, 256, 0, stream>>>(
      (const v4f*)boxes, dist, cls, (v4f*)obox, odist, ocls,
      DO_FLIP ? 1 : 0);
}